// StereoGNNSmallPretrain_15710990368915
// MI455X (gfx1250) — compile-verified
//
#include <hip/hip_runtime.h>
#include <hip/hip_bf16.h>
#include <math.h>

#define Nn 50000
#define Ee 800000
#define Gg 256
#define HD 128
#define Tt 11

typedef __attribute__((ext_vector_type(2))) float v2f;
typedef __attribute__((ext_vector_type(8))) float v8f;

// ---------- helpers ----------
__device__ __forceinline__ float wave_sum32(float v) {
  #pragma unroll
  for (int off = 16; off > 0; off >>= 1) v += __shfl_xor(v, off, 32);
  return v;
}

// order-preserving float->uint key for atomicMax-based segment max
__device__ __forceinline__ unsigned f2key(float f) {
  int i = __float_as_int(f);
  return (i >= 0) ? ((unsigned)i | 0x80000000u) : (unsigned)(~i);
}
__device__ __forceinline__ float key2f(unsigned k) {
  int i = (k & 0x80000000u) ? (int)(k & 0x7fffffffu) : ~((int)k);
  return __int_as_float(i);
}

__device__ __forceinline__ v8f wmma4(v2f a, v2f b, v8f c) {
  return __builtin_amdgcn_wmma_f32_16x16x4_f32(false, a, false, b, (short)0, c, false, false);
}

// ---------- generic fp32 WMMA GEMM: C[M,Nc] = A[M,K] @ B[K,Nc] (+ bias[col]) ----------
// one wave per 16x32 output tile (dual accumulators); M % 16 == 0, Nc % 32 == 0, any K.
// A rows must be 8-byte aligned fragments: (row*K*4 + ka*4) % 8 == 0 holds for K in {18,64,86,128}.
__global__ void gemm_wmma_f32(const float* __restrict__ A, const float* __restrict__ B,
                              const float* __restrict__ bias, float* __restrict__ C,
                              int M, int K, int Nc) {
  int wavesPerBlk = blockDim.x >> 5;
  int waveId = blockIdx.x * wavesPerBlk + (threadIdx.x >> 5);
  int tilesN = Nc >> 5;
  int tilesM = M >> 4;
  if (waveId >= tilesM * tilesN) return;            // wave-uniform exit
  int tileM = waveId / tilesN, tileN = waveId % tilesN;
  int row0 = tileM << 4, col0 = tileN << 5;
  int lane = threadIdx.x & 31;
  int m = lane & 15, kh = lane >> 4;

  v8f acc0 = {}, acc1 = {};
  const float* Arow = A + (size_t)(row0 + m) * K;
  const int K4 = K & ~3;
  for (int k0 = 0; k0 < K4; k0 += 4) {
    int ka = k0 + (kh << 1);
    v2f a = *(const v2f*)(Arow + ka);               // global_load_b64
    const float* B0 = B + (size_t)ka * Nc + col0 + m;
    const float* B1 = B + (size_t)(ka + 1) * Nc + col0 + m;
    v2f b0, b1;
    b0.x = B0[0];  b0.y = B1[0];
    b1.x = B0[16]; b1.y = B1[16];
    acc0 = wmma4(a, b0, acc0);
    acc1 = wmma4(a, b1, acc1);
  }
  if (K4 < K) {                                     // branch-free masked tail (K%4 in {1,2,3})
    int ka = K4 + (kh << 1);
    int ok0 = (ka < K), ok1 = (ka + 1 < K);
    int i0 = ok0 ? ka : 0, i1 = ok1 ? (ka + 1) : 0; // clamped safe indices
    float f0 = ok0 ? 1.f : 0.f, f1 = ok1 ? 1.f : 0.f;
    v2f a, b0, b1;
    a.x = Arow[i0] * f0;
    a.y = Arow[i1] * f1;
    const float* B0 = B + (size_t)i0 * Nc + col0 + m;
    const float* B1 = B + (size_t)i1 * Nc + col0 + m;
    b0.x = B0[0] * f0;  b0.y = B1[0] * f1;
    b1.x = B0[16] * f0; b1.y = B1[16] * f1;
    acc0 = wmma4(a, b0, acc0);
    acc1 = wmma4(a, b1, acc1);
  }
  int col = col0 + m;
  float bv0 = bias ? bias[col] : 0.f;
  float bv1 = bias ? bias[col + 16] : 0.f;
  int rbase = row0 + (kh << 3);
  #pragma unroll
  for (int r = 0; r < 8; ++r) {
    size_t rowoff = (size_t)(rbase + r) * Nc;
    C[rowoff + col]      = acc0[r] + bv0;
    C[rowoff + col + 16] = acc1[r] + bv1;
  }
}

// ---------- LayerNorm (+opt pre-bias) + ReLU (+opt residual) ; one wave per row ----------
__global__ void ln_relu_kernel(const float* __restrict__ pre, const float* __restrict__ add_bias,
                               const float* __restrict__ g, const float* __restrict__ b,
                               float* __restrict__ out, const float* __restrict__ residual,
                               int M, int D) {
  int row = blockIdx.x * (blockDim.x >> 5) + (threadIdx.x >> 5);
  int lane = threadIdx.x & 31;
  if (row >= M) return;
  const float* prow = pre + (size_t)row * D;
  int nv = D >> 5;                                  // D in {64,128}
  float vals[4];
  float s = 0.f;
  for (int i = 0; i < nv; ++i) {
    int c = lane + (i << 5);
    float v = prow[c];
    if (add_bias) v += add_bias[c];
    vals[i] = v; s += v;
  }
  float mu = wave_sum32(s) / (float)D;
  float vs = 0.f;
  for (int i = 0; i < nv; ++i) { float d = vals[i] - mu; vs += d * d; }
  float var = wave_sum32(vs) / (float)D;
  float inv = rsqrtf(var + 1e-5f);
  for (int i = 0; i < nv; ++i) {
    int c = lane + (i << 5);
    float o = (vals[i] - mu) * inv * g[c] + b[c];
    o = fmaxf(o, 0.f);
    size_t idx = (size_t)row * D + c;
    out[idx] = residual ? (o + residual[idx]) : o;
  }
}

// ---------- self-loop edge feature = mean of incoming encoded edge feats ----------
__global__ void loopea_accum(const float* __restrict__ ea, const int* __restrict__ ei,
                             float* __restrict__ lsum, float* __restrict__ cnt) {
  long long idx = (long long)blockIdx.x * blockDim.x + threadIdx.x;   // over E*64
  if (idx >= (long long)Ee * 64) return;
  int e = (int)(idx >> 6), c = (int)(idx & 63);
  int d = ei[Ee + e];
  atomicAdd(&lsum[(size_t)d * 64 + c], ea[(size_t)e * 64 + c]);
  if (c == 0) atomicAdd(&cnt[d], 1.f);
}
__global__ void loopea_final(const float* __restrict__ lsum, const float* __restrict__ cnt,
                             float* __restrict__ ea2) {
  long long idx = (long long)blockIdx.x * blockDim.x + threadIdx.x;   // over N*64
  if (idx >= (long long)Nn * 64) return;
  int i = (int)(idx >> 6), c = (int)(idx & 63);
  ea2[(size_t)(Ee + i) * 64 + c] = lsum[(size_t)i * 64 + c] / fmaxf(cnt[i], 1.f);
}

// ---------- fused edge kernel: ep GEMM (WMMA) + gather + leakyReLU + att-dot + segmax ----------
// block = 128 threads (4 waves), handles 16 edges; wave w -> cols [32w, 32w+32) of ep[16,128]
__global__ void edge_score_kernel(const float* __restrict__ ea2, const float* __restrict__ We,
                                  const float* __restrict__ xl, const float* __restrict__ xr,
                                  const float* __restrict__ att, const int* __restrict__ ei,
                                  float* __restrict__ scores, unsigned* __restrict__ smax) {
  int e0 = blockIdx.x << 4;
  int w = threadIdx.x >> 5;
  int lane = threadIdx.x & 31;
  int m = lane & 15, kh = lane >> 4;
  int col0 = w << 5;

  v8f acc0 = {}, acc1 = {};
  const float* Arow = ea2 + (size_t)(e0 + m) * 64;
  #pragma unroll
  for (int k0 = 0; k0 < 64; k0 += 4) {
    int ka = k0 + (kh << 1);
    v2f a = *(const v2f*)(Arow + ka);
    const float* B0 = We + (size_t)ka * HD + col0 + m;
    const float* B1 = We + (size_t)(ka + 1) * HD + col0 + m;
    v2f b0, b1;
    b0.x = B0[0];  b0.y = B1[0];
    b1.x = B0[16]; b1.y = B1[16];
    acc0 = wmma4(a, b0, acc0);
    acc1 = wmma4(a, b1, acc1);
  }

  __shared__ float s_sc[32];                        // [16 edges][2 heads]
  if (threadIdx.x < 32) s_sc[threadIdx.x] = 0.f;
  __syncthreads();

  int colA = col0 + m, colB = col0 + m + 16;
  int headA = colA >> 6, headB = colB >> 6;
  float avA = att[(headA << 6) + (colA & 63)];
  float avB = att[(headB << 6) + (colB & 63)];
  #pragma unroll
  for (int r = 0; r < 8; ++r) {
    int erow = (kh << 3) + r;
    int e = e0 + erow;
    int s = (e < Ee) ? ei[e]      : (e - Ee);
    int d = (e < Ee) ? ei[Ee + e] : (e - Ee);
    size_t so = (size_t)s * HD, dofs = (size_t)d * HD;
    float mvA = xl[so + colA] + xr[dofs + colA] + acc0[r];
    float mvB = xl[so + colB] + xr[dofs + colB] + acc1[r];
    mvA = (mvA > 0.f) ? mvA : 0.2f * mvA;           // leaky_relu 0.2
    mvB = (mvB > 0.f) ? mvB : 0.2f * mvB;
    atomicAdd(&s_sc[erow * 2 + headA], avA * mvA);
    atomicAdd(&s_sc[erow * 2 + headB], avB * mvB);
  }
  __syncthreads();

  if (threadIdx.x < 32) {
    int erow = threadIdx.x >> 1, hh = threadIdx.x & 1;
    int e = e0 + erow;
    float sc = s_sc[threadIdx.x];
    scores[(size_t)e * 2 + hh] = sc;
    int d = (e < Ee) ? ei[Ee + e] : (e - Ee);
    atomicMax(&smax[(size_t)d * 2 + hh], f2key(sc));
  }
}

// ---------- exp(score - segmax), accumulate denominator ----------
__global__ void softmax_norm_kernel(float* __restrict__ scores, const unsigned* __restrict__ smax,
                                    float* __restrict__ denom, const int* __restrict__ ei) {
  int idx = blockIdx.x * blockDim.x + threadIdx.x;  // over (E+N)*2
  if (idx >= (Ee + Nn) * 2) return;
  int e = idx >> 1, hh = idx & 1;
  int d = (e < Ee) ? ei[Ee + e] : (e - Ee);
  float a = __expf(scores[idx] - key2f(smax[(size_t)d * 2 + hh]));
  scores[idx] = a;
  atomicAdd(&denom[(size_t)d * 2 + hh], a);
}

// ---------- out[dst] += xl[src] * a/(denom+eps) ----------
__global__ void aggregate_kernel(const float* __restrict__ scores, const float* __restrict__ denom,
                                 const float* __restrict__ xl, const int* __restrict__ ei,
                                 float* __restrict__ hnew) {
  long long idx = (long long)blockIdx.x * blockDim.x + threadIdx.x;   // over (E+N)*128
  if (idx >= (long long)(Ee + Nn) * HD) return;
  int e = (int)(idx >> 7), col = (int)(idx & 127);
  int hh = col >> 6;
  int s = (e < Ee) ? ei[e]      : (e - Ee);
  int d = (e < Ee) ? ei[Ee + e] : (e - Ee);
  float wgt = scores[(size_t)e * 2 + hh] / (denom[(size_t)d * 2 + hh] + 1e-16f);
  atomicAdd(&hnew[(size_t)d * HD + col], xl[(size_t)s * HD + col] * wgt);
}

// ---------- global mean pool ----------
__global__ void pool_accum(const float* __restrict__ h, const int* __restrict__ batch,
                           float* __restrict__ gsum, float* __restrict__ gcnt) {
  long long idx = (long long)blockIdx.x * blockDim.x + threadIdx.x;   // over N*128
  if (idx >= (long long)Nn * HD) return;
  int i = (int)(idx >> 7), col = (int)(idx & 127);
  int b = batch[i];
  atomicAdd(&gsum[(size_t)b * HD + col], h[(size_t)i * HD + col]);
  if (col == 0) atomicAdd(&gcnt[b], 1.f);
}
__global__ void pool_final(const float* __restrict__ gsum, const float* __restrict__ gcnt,
                           float* __restrict__ graph) {
  int idx = blockIdx.x * blockDim.x + threadIdx.x;                    // over G*128
  if (idx >= Gg * HD) return;
  graph[idx] = gsum[idx] / fmaxf(gcnt[idx >> 7], 1.f);
}

__global__ void tanh_kernel(const float* __restrict__ in, float* __restrict__ out, int n) {
  int i = blockIdx.x * blockDim.x + threadIdx.x;
  if (i < n) out[i] = tanhf(in[i]);
}

// ---------- per-(t,g) 2-layer MLP heads ----------
__global__ void heads_kernel(const float* __restrict__ emb, const float* __restrict__ W1,
                             const float* __restrict__ b1, const float* __restrict__ W2,
                             const float* __restrict__ b2, float* __restrict__ out) {
  int g = blockIdx.x, t = blockIdx.y, k = threadIdx.x;                // 64 threads
  __shared__ float s_emb[HD];
  __shared__ float s_mid[64];
  s_emb[k]      = emb[(size_t)g * HD + k];
  s_emb[k + 64] = emb[(size_t)g * HD + k + 64];
  __syncthreads();
  float acc = b1[t * 64 + k];
  const float* w1 = W1 + (size_t)t * HD * 64;
  for (int d = 0; d < HD; ++d) acc += s_emb[d] * w1[(size_t)d * 64 + k];
  s_mid[k] = fmaxf(acc, 0.f);
  __syncthreads();
  if (k < 3) {
    float o = b2[t * 3 + k];
    const float* w2 = W2 + (size_t)t * 64 * 3;
    for (int d = 0; d < 64; ++d) o += s_mid[d] * w2[d * 3 + k];
    out[((size_t)t * Gg + g) * 3 + k] = o;
  }
}

static inline int cdiv(long long a, long long b) { return (int)((a + b - 1) / b); }

extern "C" void kernel_launch(void* const* d_in, const int* in_sizes, int n_in,
                              void* d_out, int out_size, void* d_ws, size_t ws_size,
                              hipStream_t stream) {
  const float* x    = (const float*)d_in[0];
  const float* eatt = (const float*)d_in[1];
  const int*   ei   = (const int*)d_in[2];
  const int*   bat  = (const int*)d_in[3];
  const float* ne_W = (const float*)d_in[4];
  const float* ne_b = (const float*)d_in[5];
  const float* ne_g = (const float*)d_in[6];
  const float* ne_be= (const float*)d_in[7];
  const float* ee_W = (const float*)d_in[8];
  const float* ee_b = (const float*)d_in[9];
  const float* ee_g = (const float*)d_in[10];
  const float* ee_be= (const float*)d_in[11];
  const float* gWl  = (const float*)d_in[12];
  const float* gbl  = (const float*)d_in[13];
  const float* gWr  = (const float*)d_in[14];
  const float* gbr  = (const float*)d_in[15];
  const float* gWe  = (const float*)d_in[16];
  const float* gatt = (const float*)d_in[17];
  const float* gbias= (const float*)d_in[18];
  const float* ln_g = (const float*)d_in[19];
  const float* ln_b = (const float*)d_in[20];
  const float* ro_W = (const float*)d_in[21];
  const float* ro_b = (const float*)d_in[22];
  const float* hW1  = (const float*)d_in[23];
  const float* hb1  = (const float*)d_in[24];
  const float* hW2  = (const float*)d_in[25];
  const float* hb2  = (const float*)d_in[26];
  float* out = (float*)d_out;

  // workspace carve-up (floats)
  float* W = (float*)d_ws;
  size_t o = 0;
  float* h      = W + o; o += (size_t)Nn * HD;
  float* tmp    = W + o; o += (size_t)Nn * HD;          // gemm pre / h_new accumulator
  float* xl     = W + o; o += (size_t)Nn * HD;
  float* xr     = W + o; o += (size_t)Nn * HD;
  float* ea2    = W + o; o += (size_t)(Ee + Nn) * 64;
  float* scores = W + o; o += (size_t)(Ee + Nn) * 2;
  unsigned* smax = (unsigned*)(W + o); o += (size_t)Nn * 2;
  float* denom  = W + o; o += (size_t)Nn * 2;
  float* cnt    = W + o; o += (size_t)Nn;
  float* lsum   = W + o; o += (size_t)Nn * 64;
  float* gsum   = W + o; o += (size_t)Gg * HD;
  float* gcnt   = W + o; o += (size_t)Gg;
  float* graph  = W + o; o += (size_t)Gg * HD;
  float* embp   = W + o; o += (size_t)Gg * HD;
  float* emb    = W + o; o += (size_t)Gg * HD;
  (void)ws_size; (void)n_in; (void)in_sizes; (void)out_size;

  const int TB = 256;
  auto gemm_blocks = [](int M, int Nc) { return cdiv((long long)(M >> 4) * (Nc >> 5), 8); };

  // 1) node encoder: h = relu(LN(x @ ne_W + ne_b))
  gemm_wmma_f32<<<gemm_blocks(Nn, HD), TB, 0, stream>>>(x, ne_W, ne_b, tmp, Nn, 86, HD);
  ln_relu_kernel<<<cdiv(Nn, 8), TB, 0, stream>>>(tmp, nullptr, ne_g, ne_be, h, nullptr, Nn, HD);

  // 2) edge encoder: ea2[0:E] = relu(LN(edge_attr @ ee_W + ee_b))
  gemm_wmma_f32<<<gemm_blocks(Ee, 64), TB, 0, stream>>>(eatt, ee_W, ee_b, ea2, Ee, 18, 64);
  ln_relu_kernel<<<cdiv(Ee, 8), TB, 0, stream>>>(ea2, nullptr, ee_g, ee_be, ea2, nullptr, Ee, 64);

  // 3) self-loop edge features (mean of incoming)
  hipMemsetAsync(cnt, 0, sizeof(float) * Nn, stream);
  hipMemsetAsync(lsum, 0, sizeof(float) * (size_t)Nn * 64, stream);
  loopea_accum<<<cdiv((long long)Ee * 64, TB), TB, 0, stream>>>(ea2, ei, lsum, cnt);
  loopea_final<<<cdiv((long long)Nn * 64, TB), TB, 0, stream>>>(lsum, cnt, ea2);

  // 4) GATv2 layers
  for (int l = 0; l < 2; ++l) {
    gemm_wmma_f32<<<gemm_blocks(Nn, HD), TB, 0, stream>>>(h, gWl + (size_t)l * HD * HD,
                                                          gbl + (size_t)l * HD, xl, Nn, HD, HD);
    gemm_wmma_f32<<<gemm_blocks(Nn, HD), TB, 0, stream>>>(h, gWr + (size_t)l * HD * HD,
                                                          gbr + (size_t)l * HD, xr, Nn, HD, HD);
    hipMemsetAsync(smax, 0, sizeof(unsigned) * (size_t)Nn * 2, stream);   // key 0 == most-negative
    hipMemsetAsync(denom, 0, sizeof(float) * (size_t)Nn * 2, stream);
    edge_score_kernel<<<(Ee + Nn) / 16, 128, 0, stream>>>(
        ea2, gWe + (size_t)l * 64 * HD, xl, xr, gatt + (size_t)l * HD, ei, scores, smax);
    softmax_norm_kernel<<<cdiv((long long)(Ee + Nn) * 2, TB), TB, 0, stream>>>(scores, smax, denom, ei);
    hipMemsetAsync(tmp, 0, sizeof(float) * (size_t)Nn * HD, stream);
    aggregate_kernel<<<cdiv((long long)(Ee + Nn) * HD, TB), TB, 0, stream>>>(scores, denom, xl, ei, tmp);
    // h = relu(LN(h_new + gat_bias)) + h
    ln_relu_kernel<<<cdiv(Nn, 8), TB, 0, stream>>>(tmp, gbias + (size_t)l * HD,
                                                   ln_g + (size_t)l * HD, ln_b + (size_t)l * HD,
                                                   h, h, Nn, HD);
  }

  // 5) global mean pool
  hipMemsetAsync(gsum, 0, sizeof(float) * (size_t)Gg * HD, stream);
  hipMemsetAsync(gcnt, 0, sizeof(float) * (size_t)Gg, stream);
  pool_accum<<<cdiv((long long)Nn * HD, TB), TB, 0, stream>>>(h, bat, gsum, gcnt);
  pool_final<<<cdiv(Gg * HD, TB), TB, 0, stream>>>(gsum, gcnt, graph);

  // 6) readout + heads
  gemm_wmma_f32<<<gemm_blocks(Gg, HD), TB, 0, stream>>>(graph, ro_W, ro_b, embp, Gg, HD, HD);
  tanh_kernel<<<cdiv(Gg * HD, TB), TB, 0, stream>>>(embp, emb, Gg * HD);
  heads_kernel<<<dim3(Gg, Tt), 64, 0, stream>>>(emb, hW1, hb1, hW2, hb2, out);
}